// LabelDependencySmoothing_48034914238716
// MI455X (gfx1250) — compile-verified
//
#include <hip/hip_runtime.h>
#include <hip/hip_bf16.h>
#include <math.h>

// Problem constants from the reference
#define NUM_LABELS     2000
#define LAMBDA_SMOOTH  0.1

// Tuning
#define ROWS_PER_BLOCK 8      // batch rows staged in LDS per workgroup (8*2000*4 = 64000 B)
#define THREADS        256    // 8 waves (wave32)
#define EDGE_SLICES    4      // grid.y: split edge list for occupancy

typedef float v2f __attribute__((ext_vector_type(2)));
typedef float v8f __attribute__((ext_vector_type(8)));

// Stage 1: each block stages ROWS_PER_BLOCK rows of y_converted in LDS,
// sweeps a slice of the edge list, and writes one partial sum.
__global__ __launch_bounds__(THREADS)
void lds_edge_partial_kernel(const float* __restrict__ logits,
                             const float* __restrict__ labels,
                             const float* __restrict__ ew,
                             const int*   __restrict__ lidx,
                             const int*   __restrict__ ridx,
                             float*       __restrict__ partials,
                             int batch, int nedges)
{
    __shared__ float sY[ROWS_PER_BLOCK * NUM_LABELS]; // 64000 B
    __shared__ float sRed[THREADS];                   // 1024 B

    const int tid = threadIdx.x;
    const int b0  = blockIdx.x * ROWS_PER_BLOCK;

    // Build y_converted for our rows: y = has_label ? 2*label-1 : tanh(logit/2)
    // (2*sigmoid(x)-1 == tanh(x/2); labels are exactly {-1,0,1} so lab>=0 <=> annotated)
    #pragma unroll
    for (int t = 0; t < ROWS_PER_BLOCK; ++t) {
        const int b = b0 + t;
        if (b < batch) {
            const float* lg = logits + (size_t)b * NUM_LABELS;
            const float* lb = labels + (size_t)b * NUM_LABELS;
            for (int c = tid; c < NUM_LABELS; c += THREADS) {
                const float lab = lb[c];
                float y;
                if (lab >= 0.0f) y = 2.0f * lab - 1.0f;
                else             y = tanhf(0.5f * lg[c]);
                sY[t * NUM_LABELS + c] = y;
            }
        } else {
            for (int c = tid; c < NUM_LABELS; c += THREADS)
                sY[t * NUM_LABELS + c] = 0.0f;   // zero rows contribute 0 to diff^2
        }
    }
    __syncthreads();

    // Sweep our edge slice; gather both endpoints from LDS for all 8 rows.
    float acc = 0.0f;
    const int estart  = blockIdx.y * THREADS + tid;
    const int estride = gridDim.y * THREADS;
    for (int e = estart; e < nedges; e += estride) {
        const int   l = lidx[e];
        const int   r = ridx[e];
        const float w = ew[e];
        float s = 0.0f;
        #pragma unroll
        for (int t = 0; t < ROWS_PER_BLOCK; ++t) {
            const float d = sY[t * NUM_LABELS + l] - sY[t * NUM_LABELS + r];
            s = fmaf(d, d, s);
        }
        acc = fmaf(w, s, acc);
    }

    // Deterministic block reduction.
    sRed[tid] = acc;
    __syncthreads();
    #pragma unroll
    for (int off = THREADS / 2; off > 0; off >>= 1) {
        if (tid < off) sRed[tid] += sRed[tid + off];
        __syncthreads();
    }
    if (tid == 0)
        partials[blockIdx.y * gridDim.x + blockIdx.x] = sRed[0];
}

// Stage 2: single wave32. Lane-strided sum of partials, then a
// V_WMMA_F32_16X16X4_F32 cross-lane reduction (A = lane sums, B = ones:
// D[m,n] = sum_k A[m,k]), folded with two readlanes. EXEC is all ones
// (exactly one full wave, no divergence before the WMMA).
__global__ __launch_bounds__(32)
void final_reduce_wmma_kernel(const float* __restrict__ partials,
                              int nparts, float scale,
                              float* __restrict__ out)
{
    const int lane = threadIdx.x;                 // 0..31, full wave
    float laneSum = 0.0f;
    for (int i = lane; i < nparts; i += 32)
        laneSum += partials[i];

    // A (16x4 f32): lane L<16 holds A[L, 0..1], lane L+16 holds A[L, 2..3].
    // Put laneSum in the first slot, zero the second => rowsum[m] = lane m + lane m+16.
    v2f a; a[0] = laneSum; a[1] = 0.0f;
    v2f b; b[0] = 1.0f;    b[1] = 1.0f;           // all-ones B (4x16) — layout-invariant
    v8f c = {};
    v8f d = __builtin_amdgcn_wmma_f32_16x16x4_f32(
        /*neg_a=*/false, a, /*neg_b=*/false, b,
        /*c_mod=*/(short)0, c, /*reuse_a=*/false, /*reuse_b=*/false);

    // D layout: VGPR r, lanes 0-15 -> M=r, lanes 16-31 -> M=8+r.
    // Per-lane sum of the 8 D regs gives: lanes 0-15 = sum rowsum[0..7],
    // lanes 16-31 = sum rowsum[8..15]. Total = lane0 + lane16.
    float s = 0.0f;
    #pragma unroll
    for (int i = 0; i < 8; ++i) s += d[i];

    const float total =
        __int_as_float(__builtin_amdgcn_readlane(__float_as_int(s), 0)) +
        __int_as_float(__builtin_amdgcn_readlane(__float_as_int(s), 16));

    if (lane == 0)
        out[0] = scale * total;
}

extern "C" void kernel_launch(void* const* d_in, const int* in_sizes, int n_in,
                              void* d_out, int out_size, void* d_ws, size_t ws_size,
                              hipStream_t stream)
{
    const float* logits = (const float*)d_in[0];
    const float* labels = (const float*)d_in[1];
    const float* ew     = (const float*)d_in[2];
    const int*   lidx   = (const int*)d_in[3];
    const int*   ridx   = (const int*)d_in[4];

    const int batch  = in_sizes[0] / NUM_LABELS;   // 1024
    const int nedges = in_sizes[2];                // ~40K

    float* partials = (float*)d_ws;

    const int gx = (batch + ROWS_PER_BLOCK - 1) / ROWS_PER_BLOCK;  // 128
    dim3 grid(gx, EDGE_SLICES);
    lds_edge_partial_kernel<<<grid, THREADS, 0, stream>>>(
        logits, labels, ew, lidx, ridx, partials, batch, nedges);

    const int nparts = gx * EDGE_SLICES;           // 512
    const float scale =
        (float)(LAMBDA_SMOOTH / ((double)batch * (double)nedges));
    final_reduce_wmma_kernel<<<1, 32, 0, stream>>>(
        partials, nparts, scale, (float*)d_out);
}